// HomoGNNModel_32384053412201
// MI455X (gfx1250) — compile-verified
//
#include <hip/hip_runtime.h>

typedef float v2f __attribute__((ext_vector_type(2)));
typedef float v4f __attribute__((ext_vector_type(4)));
typedef float v8f __attribute__((ext_vector_type(8)));

#define D_FEAT   256      // D_IN == D_HID == 256
#define N1_ROWS  11264
#define N2_ROWS  1024
#define FAN0     25
#define FAN1     10
#define NCLASS   128

// ---------------------------------------------------------------------------
// Fused gather + mean:  out[row, :] = mean_k src[ map(col[row*fanout+k]), : ]
// ONE WAVE (32 lanes) per destination row, launched as 1-wave workgroups so
// `row` is derived purely from blockIdx -> the col[] / gid[] index chain is
// wave-uniform and lowers to scalar s_load (KMcnt), keeping the vector memory
// pipe free for the row payload. Each lane accumulates two float4 chunks:
// 32 lanes x 2 x 4 floats = 256-float row, i.e. two global_load_b128 per
// neighbor instead of 8 waves x b32 — 4x fewer VMEM ops on the path that
// carries ~290MB of HBM traffic (the roofline limiter at 23.3 TB/s).
// ---------------------------------------------------------------------------
__global__ void gather_mean_kernel(const float* __restrict__ src,
                                   const int*   __restrict__ col,
                                   const int*   __restrict__ gid,
                                   float*       __restrict__ out,
                                   int fanout, float inv_fanout, int use_gid)
{
    const int row  = blockIdx.x;
    const int j    = threadIdx.x * 4;       // lane's float4 slot in first half
    const int* cptr = col + (long long)row * fanout;

    v4f s0 = {0.0f, 0.0f, 0.0f, 0.0f};
    v4f s1 = {0.0f, 0.0f, 0.0f, 0.0f};
    for (int k = 0; k < fanout; ++k) {
        int idx  = cptr[k];                 // uniform -> scalar load
        int srow = use_gid ? gid[idx] : idx;// uniform -> scalar load
        const float* base = src + (long long)srow * D_FEAT;
        s0 += *(const v4f*)(base + j);
        s1 += *(const v4f*)(base + 128 + j);
    }
    float* obase = out + (long long)row * D_FEAT;
    *(v4f*)(obase + j)       = s0 * inv_fanout;
    *(v4f*)(obase + 128 + j) = s1 * inv_fanout;
}

// ---------------------------------------------------------------------------
// Dual-operand SAGE GEMM via fp32 WMMA:
//   out = act( Aself @ Wself + Anb @ Wneigh + bias )
// One wave computes one 16x16 output tile. K = 256, stepped by 4 with
// V_WMMA_F32_16X16X4_F32. A-fragment (16x4 f32): lanes 0-15 carry M=0..15
// with {v0,v1} = {K+0,K+1}; lanes 16-31 carry {K+2,K+3}. B-fragment (4x16)
// mirrored. C/D: 8 VGPRs, M = r + 8*(lane>=16), N = lane&15.
// USE_GID: self-term rows are gathered through gid (emb[gid0[row]]), which
// avoids materializing the 300MB x = emb[gid0] tensor. W matrices (256KB)
// stay L2-resident, so per-wave B refetch is cheap.
// ---------------------------------------------------------------------------
template<int NCOL, bool USE_GID, bool RELU>
__global__ void sage_gemm_kernel(const float* __restrict__ Aself,
                                 const int*   __restrict__ gid,
                                 const float* __restrict__ Anb,
                                 const float* __restrict__ Wself,
                                 const float* __restrict__ Wneigh,
                                 const float* __restrict__ bias,
                                 float*       __restrict__ out)
{
    constexpr int K  = 256;
    constexpr int TN = NCOL / 16;           // tiles along N

    const int lane = threadIdx.x & 31;
    const int wave = threadIdx.x >> 5;
    const int tile = blockIdx.x * 8 + wave; // 8 waves per block, exact fit
    const int tm   = (tile / TN) * 16;
    const int tn   = (tile % TN) * 16;
    const int ml   = lane & 15;             // M for A rows, N for B/C cols
    const int hi   = lane >> 4;             // 0: K{0,1}, 1: K{2,3}

    const int arow = tm + ml;
    long long aoff_self = USE_GID ? (long long)gid[arow] * K
                                  : (long long)arow * K;
    const long long aoff_nb = (long long)arow * K;
    const float bv = bias[tn + ml];         // hoisted: overlaps with K-loop

    v8f c = {};
#pragma unroll 4
    for (int k = 0; k < K; k += 4) {
        const int ka = k + 2 * hi;          // even -> 8B aligned v2f loads
        v2f a_s = *(const v2f*)(Aself + aoff_self + ka);
        v2f a_n = *(const v2f*)(Anb   + aoff_nb   + ka);
        v2f b_s, b_n;
        b_s.x = Wself [(long long)(ka    ) * NCOL + tn + ml];
        b_s.y = Wself [(long long)(ka + 1) * NCOL + tn + ml];
        b_n.x = Wneigh[(long long)(ka    ) * NCOL + tn + ml];
        b_n.y = Wneigh[(long long)(ka + 1) * NCOL + tn + ml];
        c = __builtin_amdgcn_wmma_f32_16x16x4_f32(false, a_s, false, b_s,
                                                  (short)0, c, false, false);
        c = __builtin_amdgcn_wmma_f32_16x16x4_f32(false, a_n, false, b_n,
                                                  (short)0, c, false, false);
    }

#pragma unroll
    for (int r = 0; r < 8; ++r) {
        float v = c[r] + bv;
        if (RELU) v = v > 0.0f ? v : 0.0f;
        out[(long long)(tm + r + 8 * hi) * NCOL + tn + ml] = v;
    }
}

// ---------------------------------------------------------------------------
// Launch: inputs in setup_inputs() order:
// 0 emb, 1 gid0, 2 col0, 3 col1, 4 W_self0, 5 W_neigh0, 6 b0,
// 7 W_self1, 8 W_neigh1, 9 b1.  Output: 1024 x 128 fp32.
// Workspace: nb0 (11264x256f) | h (11264x256f) | nb1 (1024x256f) ~ 24MB.
// ---------------------------------------------------------------------------
extern "C" void kernel_launch(void* const* d_in, const int* in_sizes, int n_in,
                              void* d_out, int out_size, void* d_ws, size_t ws_size,
                              hipStream_t stream)
{
    const float* emb     = (const float*)d_in[0];
    const int*   gid0    = (const int*)  d_in[1];
    const int*   col0    = (const int*)  d_in[2];
    const int*   col1    = (const int*)  d_in[3];
    const float* Wself0  = (const float*)d_in[4];
    const float* Wneigh0 = (const float*)d_in[5];
    const float* b0      = (const float*)d_in[6];
    const float* Wself1  = (const float*)d_in[7];
    const float* Wneigh1 = (const float*)d_in[8];
    const float* b1      = (const float*)d_in[9];
    float* out = (float*)d_out;

    char*  ws  = (char*)d_ws;
    const size_t rowbytes = (size_t)D_FEAT * sizeof(float);
    float* nb0 = (float*)ws;
    float* h   = (float*)(ws + (size_t)N1_ROWS * rowbytes);
    float* nb1 = (float*)(ws + 2 * (size_t)N1_ROWS * rowbytes);

    // Layer 0: neighbor mean over emb[gid0[col0]] (dominant HBM traffic ~288MB)
    gather_mean_kernel<<<N1_ROWS, 32, 0, stream>>>(
        emb, col0, gid0, nb0, FAN0, 1.0f / FAN0, 1);

    // Layer 0 GEMM: h = relu(x_tgt @ Wself0 + nb0 @ Wneigh0 + b0)
    const int tiles0 = (N1_ROWS / 16) * (D_FEAT / 16);  // 11264 tiles
    sage_gemm_kernel<D_FEAT, true, true><<<tiles0 / 8, 256, 0, stream>>>(
        emb, gid0, nb0, Wself0, Wneigh0, b0, h);

    // Layer 1: neighbor mean over h[col1]
    gather_mean_kernel<<<N2_ROWS, 32, 0, stream>>>(
        h, col1, nullptr, nb1, FAN1, 1.0f / FAN1, 0);

    // Layer 1 GEMM: out = h[:1024] @ Wself1 + nb1 @ Wneigh1 + b1
    const int tiles1 = (N2_ROWS / 16) * (NCLASS / 16);  // 512 tiles
    sage_gemm_kernel<NCLASS, false, false><<<tiles1 / 8, 256, 0, stream>>>(
        h, nullptr, nb1, Wself1, Wneigh1, b1, out);
}